// TtLlamaAttention_52604759441678
// MI455X (gfx1250) — compile-verified
//
#include <hip/hip_runtime.h>
#include <hip/hip_bf16.h>

typedef __attribute__((ext_vector_type(2))) float v2f;
typedef __attribute__((ext_vector_type(8))) float v8f;

#define BATCH   32
#define HIDDEN  4096
#define NHEADS  32
#define NKV     8
#define GROUP   4
#define HD      128
#define SEQ     2048

#define KC      64      // K-chunk staged in LDS (floats)
#define ASTRIDE 68      // padded LDS row stride (floats): kills bank conflicts,
                        // keeps 16B alignment (68*4 = 272 = 16*17)

// ---------------------------------------------------------------------------
// CDNA5 async copy: global -> LDS, 16 bytes per lane, ASYNCcnt-tracked.
// ---------------------------------------------------------------------------
__device__ __forceinline__ void async_load_b128(unsigned int lds_addr,
                                                const void* gaddr) {
    asm volatile("global_load_async_to_lds_b128 %0, %1, off"
                 :: "v"(lds_addr), "v"(gaddr) : "memory");
}

// ---------------------------------------------------------------------------
// fp32 GEMM via V_WMMA_F32_16X16X4_F32 with async-LDS double buffering.
//   block = 128 threads (4 waves). All waves share the 16 A-rows (tileM);
//   the 16xKC A-chunk is staged into LDS with global_load_async_to_lds_b128
//   (2 async ops per wave per chunk -> s_wait_asynccnt 2 pipelines chunk c+1
//   behind chunk c). Each wave computes a 16x32 output slab (2 WMMA tiles).
// ISA layouts (CDNA5 ISA §7.12.2):
//   A 16x4 : lane = row(0..15) | half*16 ; vgpr0=K(2*half), vgpr1=K(2*half+1)
//   B 4x16 : lane = col(0..15) | half*16 ; vgpr0=K(2*half), vgpr1=K(2*half+1)
//   C 16x16: vgpr v -> row = v + half*8, col = lane&15
// Requires: M%16==0, N a multiple of 128 (grid.x=N/128), K%64==0.
// ---------------------------------------------------------------------------
__global__ __launch_bounds__(128)
void gemm_wmma_f32(const float* __restrict__ A, const float* __restrict__ B,
                   float* __restrict__ C, int M, int K, int N) {
    __shared__ float As[2][16][ASTRIDE];

    const int tid    = threadIdx.x;        // 0..127
    const int waveId = tid >> 5;           // 0..3
    const int lane   = tid & 31;
    const int half   = lane >> 4;          // 0|1
    const int lane16 = lane & 15;

    const int tileM  = blockIdx.y * 16;
    const int tileN  = blockIdx.x * 128 + waveId * 32;

    // --- A staging: thread copies rows r0 and r0+8, one 16B segment each ---
    const int r0 = tid >> 4;               // 0..7
    const int ac = (tid & 15) * 4;         // col (floats) within chunk
    const char* ag0 = (const char*)(A + (size_t)(tileM + r0) * K + ac);
    const char* ag1 = ag0 + (size_t)8 * K * sizeof(float);
    const unsigned int l0 = (unsigned int)(unsigned long long)&As[0][r0][ac];
    const unsigned int l1 = (unsigned int)(unsigned long long)&As[0][r0 + 8][ac];
    const unsigned int lbuf = (unsigned int)(sizeof(float) * 16 * ASTRIDE);

    // --- B stream: two row pointers (k+2*half) and (k+2*half+1), add-only ---
    const size_t rowN = (size_t)N;
    const float* bp0 = B + (size_t)(half * 2) * rowN + tileN + lane16;
    const float* bp1 = bp0 + rowN;

    v8f acc0 = {}, acc1 = {};

    const int nchunks = K / KC;
    // prologue: chunk 0 -> buffer 0
    async_load_b128(l0, ag0);
    async_load_b128(l1, ag1);

    for (int c = 0; c < nchunks; ++c) {
        const int buf = c & 1;
        if (c + 1 < nchunks) {
            const char* g0 = ag0 + (size_t)(c + 1) * (KC * sizeof(float));
            const char* g1 = ag1 + (size_t)(c + 1) * (KC * sizeof(float));
            async_load_b128(l0 + (buf ^ 1) * lbuf, g0);
            async_load_b128(l1 + (buf ^ 1) * lbuf, g1);
            asm volatile("s_wait_asynccnt 0x2" ::: "memory");  // chunk c done
        } else {
            asm volatile("s_wait_asynccnt 0x0" ::: "memory");
        }
        __syncthreads();                       // As[buf] visible to all waves

        // prefetch the weight stream one chunk ahead (global_prefetch_b8)
        __builtin_prefetch(bp0 + (size_t)KC * rowN, 0, 1);

        const float* arow = &As[buf][lane16][2 * half];
#pragma unroll 4
        for (int kk = 0; kk < KC; kk += 4) {
            v2f a = *(const v2f*)(arow + kk);  // ds_load_b64
            v2f b0, b1;
            b0.x = bp0[0];  b0.y = bp1[0];
            b1.x = bp0[16]; b1.y = bp1[16];
            bp0 += 4 * rowN; bp1 += 4 * rowN;
            acc0 = __builtin_amdgcn_wmma_f32_16x16x4_f32(false, a, false, b0,
                                                         (short)0, acc0, false, false);
            acc1 = __builtin_amdgcn_wmma_f32_16x16x4_f32(false, a, false, b1,
                                                         (short)0, acc1, false, false);
        }
        __syncthreads();   // everyone done reading As[buf] before it's refilled
    }

#pragma unroll
    for (int v = 0; v < 8; ++v) {
        float* crow = C + (size_t)(tileM + v + half * 8) * N + tileN + lane16;
        crow[0]  = acc0[v];
        crow[16] = acc1[v];
    }
}

// ---------------------------------------------------------------------------
// GQA flash-decode attention.
// grid = BATCH*NKV blocks, 256 threads (8 waves). Wave w handles s = w, w+8,...
// Per lane: float4 slice of head_dim. Online softmax per head; row `pos` of
// the KV cache is overridden with the fresh k/v (inputs are never mutated).
// ---------------------------------------------------------------------------
__global__ __launch_bounds__(256)
void attn_decode(const float* __restrict__ qrot,   // [B, NHEADS, HD]
                 const float* __restrict__ krot,   // [B, NKV, HD]
                 const float* __restrict__ vproj,  // [B, NKV, HD]
                 const float* __restrict__ keys,   // [B, NKV, SEQ, HD]
                 const float* __restrict__ values, // [B, NKV, SEQ, HD]
                 const float* __restrict__ mask,   // [B, NHEADS, 1, SEQ]
                 const int*   __restrict__ pos_ptr,
                 float* __restrict__ out) {        // [B, NHEADS, HD]
    const int b    = blockIdx.x >> 3;
    const int kv   = blockIdx.x & 7;
    const int pos  = *pos_ptr;
    const int tid  = threadIdx.x;
    const int wave = tid >> 5;   // 0..7
    const int lane = tid & 31;

    const float scale = 0.08838834764831845f;   // 1/sqrt(128)

    const float* qb = qrot + (size_t)(b * NHEADS + kv * GROUP) * HD + lane * 4;
    float4 qreg[GROUP];
#pragma unroll
    for (int g = 0; g < GROUP; ++g)
        qreg[g] = *(const float4*)(qb + g * HD);

    const float* Kb = keys   + ((size_t)(b * NKV + kv)) * SEQ * HD;
    const float* Vb = values + ((size_t)(b * NKV + kv)) * SEQ * HD;
    const float* Mb = mask   + ((size_t)(b * NHEADS + kv * GROUP)) * SEQ;
    const float* kr = krot   + (size_t)(b * NKV + kv) * HD + lane * 4;
    const float* vr = vproj  + (size_t)(b * NKV + kv) * HD + lane * 4;

    float  m[GROUP], l[GROUP];
    float4 oacc[GROUP];
#pragma unroll
    for (int g = 0; g < GROUP; ++g) {
        m[g] = -3.0e38f; l[g] = 0.0f;
        oacc[g] = make_float4(0.f, 0.f, 0.f, 0.f);
    }

    for (int s = wave; s < SEQ; s += 8) {
        const bool cur = (s == pos);
        float4 k4 = cur ? *(const float4*)kr
                        : *(const float4*)(Kb + (size_t)s * HD + lane * 4);
        float4 v4 = cur ? *(const float4*)vr
                        : *(const float4*)(Vb + (size_t)s * HD + lane * 4);

        float sc[GROUP];
#pragma unroll
        for (int g = 0; g < GROUP; ++g) {
            float p = qreg[g].x * k4.x + qreg[g].y * k4.y +
                      qreg[g].z * k4.z + qreg[g].w * k4.w;
#pragma unroll
            for (int off = 16; off > 0; off >>= 1)
                p += __shfl_xor(p, off, 32);
            sc[g] = p * scale + Mb[g * SEQ + s];
        }

#pragma unroll
        for (int g = 0; g < GROUP; ++g) {
            float mn   = fmaxf(m[g], sc[g]);
            float corr = __expf(m[g] - mn);
            float p    = __expf(sc[g] - mn);
            m[g] = mn;
            l[g] = l[g] * corr + p;
            oacc[g].x = oacc[g].x * corr + p * v4.x;
            oacc[g].y = oacc[g].y * corr + p * v4.y;
            oacc[g].z = oacc[g].z * corr + p * v4.z;
            oacc[g].w = oacc[g].w * corr + p * v4.w;
        }
    }

    // cross-wave combine through LDS
    __shared__ float so[8][GROUP][HD];
    __shared__ float sm[8][GROUP];
    __shared__ float sl[8][GROUP];
#pragma unroll
    for (int g = 0; g < GROUP; ++g) {
        *(float4*)&so[wave][g][lane * 4] = oacc[g];
        if (lane == 0) { sm[wave][g] = m[g]; sl[wave][g] = l[g]; }
    }
    __syncthreads();

    if (wave < GROUP) {
        const int g = wave;
        float M = -3.0e38f;
#pragma unroll
        for (int w = 0; w < 8; ++w) M = fmaxf(M, sm[w][g]);
        float  L = 0.0f;
        float4 O = make_float4(0.f, 0.f, 0.f, 0.f);
#pragma unroll
        for (int w = 0; w < 8; ++w) {
            float e = __expf(sm[w][g] - M);
            L += sl[w][g] * e;
            float4 t = *(float4*)&so[w][g][lane * 4];
            O.x += e * t.x; O.y += e * t.y; O.z += e * t.z; O.w += e * t.w;
        }
        float inv = 1.0f / L;
        O.x *= inv; O.y *= inv; O.z *= inv; O.w *= inv;
        *(float4*)(out + (size_t)(b * NHEADS + kv * GROUP + g) * HD + lane * 4) = O;
    }
}

// ---------------------------------------------------------------------------
extern "C" void kernel_launch(void* const* d_in, const int* in_sizes, int n_in,
                              void* d_out, int out_size, void* d_ws, size_t ws_size,
                              hipStream_t stream) {
    const float* x    = (const float*)d_in[0];
    const float* wq   = (const float*)d_in[1];
    const float* wk   = (const float*)d_in[2];
    const float* wv   = (const float*)d_in[3];
    const float* wo   = (const float*)d_in[4];
    const float* rot  = (const float*)d_in[5];
    const float* keys = (const float*)d_in[6];
    const float* vals = (const float*)d_in[7];
    const float* mask = (const float*)d_in[8];
    const int*   pos  = (const int*)d_in[9];
    float* out = (float*)d_out;

    float* ws    = (float*)d_ws;
    float* qproj = ws;                              // 32*4096
    float* kproj = qproj + BATCH * HIDDEN;          // 32*1024
    float* vproj = kproj + BATCH * NKV * HD;        // 32*1024
    float* qrot  = vproj + BATCH * NKV * HD;        // 32*4096
    float* krot  = qrot  + BATCH * HIDDEN;          // 32*1024
    float* aout  = krot  + BATCH * NKV * HD;        // 32*4096

    dim3 blk(128);
    // QKV projections
    gemm_wmma_f32<<<dim3(HIDDEN / 128, BATCH / 16), blk, 0, stream>>>(
        x, wq, qproj, BATCH, HIDDEN, HIDDEN);
    gemm_wmma_f32<<<dim3((NKV * HD) / 128, BATCH / 16), blk, 0, stream>>>(
        x, wk, kproj, BATCH, HIDDEN, NKV * HD);
    gemm_wmma_f32<<<dim3((NKV * HD) / 128, BATCH / 16), blk, 0, stream>>>(
        x, wv, vproj, BATCH, HIDDEN, NKV * HD);
    // RoPE as GEMM with rotation matrix
    gemm_wmma_f32<<<dim3(HD / 128, (BATCH * NHEADS) / 16), blk, 0, stream>>>(
        qproj, rot, qrot, BATCH * NHEADS, HD, HD);
    gemm_wmma_f32<<<dim3(HD / 128, (BATCH * NKV) / 16), blk, 0, stream>>>(
        kproj, rot, krot, BATCH * NKV, HD, HD);
    // GQA flash-decode attention
    attn_decode<<<dim3(BATCH * NKV), dim3(256), 0, stream>>>(
        qrot, krot, vproj, keys, vals, mask, pos, aout);
    // Output projection
    gemm_wmma_f32<<<dim3(HIDDEN / 128, BATCH / 16), blk, 0, stream>>>(
        aout, wo, out, BATCH, HIDDEN, HIDDEN);
}